// TitansLayer_10840497455731
// MI455X (gfx1250) — compile-verified
//
#include <hip/hip_runtime.h>
#include <hip/hip_bf16.h>
#include <math.h>

// ---------------------------------------------------------------------------
// Titans delta-rule memory layer for MI455X (gfx1250, wave32).
//   B=256, T=32, D=256, FF=1024, all fp32.
// Strategy:
//   * All state-independent projections -> V_WMMA_F32_16X16X4_F32 GEMMs;
//     each wave computes a 16x64 C strip (4 accumulators). Inner loop is
//     software-pipelined (double-buffered fragments, peeled epilogue) so
//     WMMAs only wait on loads issued a full iteration earlier.
//   * Serial scan: 1 workgroup per batch element, 256x256 fp32 state resident
//     in LDS (257-float row stride => bank-conflict free), T=32 steps.
//     state0 (64 MB, the dominant HBM stream) is pulled in via the gfx1250
//     async global->LDS DMA path when available (ASYNCcnt-tracked).
//   * Memory read (state @ q) fused into the scan kernel (state never leaves LDS).
// ---------------------------------------------------------------------------

#define Bq 256
#define Tq 32
#define Dq 256
#define FFq 1024

typedef __attribute__((ext_vector_type(2))) float v2f;
typedef __attribute__((ext_vector_type(8))) float v8f;

#if defined(__has_builtin)
#if __has_builtin(__builtin_amdgcn_global_load_async_to_lds_b32)
#define TITANS_ASYNC_LDS 1
#endif
#endif

#ifdef TITANS_ASYNC_LDS
typedef __attribute__((address_space(1))) int as1_int;
typedef __attribute__((address_space(3))) int as3_int;
#define TITANS_ASYNC_CP_B32(gsrc, ldst)                                        \
    __builtin_amdgcn_global_load_async_to_lds_b32(                             \
        (as1_int*)(void*)(gsrc), (as3_int*)(void*)(ldst), 0, 0)
#if __has_builtin(__builtin_amdgcn_s_wait_asynccnt)
#define TITANS_WAIT_ASYNC() __builtin_amdgcn_s_wait_asynccnt(0)
#else
#define TITANS_WAIT_ASYNC() asm volatile("s_wait_asynccnt 0x0" ::: "memory")
#endif
#endif

// ---------------------------------------------------------------------------
// fp32 WMMA GEMM:  C[M,N] = A[M,K] @ B[N,K]^T (+ bias[n]) (+ add[m,n]) (relu?)
// Block: 256 threads = 8 waves; wave w -> rows [(by*8+w)*16, +16),
// cols [bx*64, +64) as 4 WMMA accumulators sharing one A fragment.
// Requires M % 128 == 0, N % 64 == 0, K % 4 == 0, K >= 8.
// ---------------------------------------------------------------------------
__global__ __launch_bounds__(256)
void titans_gemm_nt_f32_wmma(const float* __restrict__ A,
                             const float* __restrict__ Bm,
                             float* __restrict__ C,
                             const float* __restrict__ bias,   // [N] or null
                             const float* __restrict__ add,    // [M,N] or null
                             int M, int N, int K, int relu)
{
    const int lane = threadIdx.x & 31;
    const int wave = threadIdx.x >> 5;       // 0..7
    const int half = lane >> 4;              // 0: lanes 0-15, 1: lanes 16-31
    const int l16  = lane & 15;
    const int m_base = (blockIdx.y * 8 + wave) * 16;
    const int n_base = blockIdx.x * 64;

    v8f acc[4];
#pragma unroll
    for (int nt = 0; nt < 4; ++nt)
        acc[nt] = (v8f){0.f, 0.f, 0.f, 0.f, 0.f, 0.f, 0.f, 0.f};

    // A 16x4 tile layout (ISA 7.12.2): lanes 0-15 hold K=k0,k0+1 ; lanes 16-31 K=k0+2,k0+3
    const float* arow = A  + (size_t)(m_base + l16) * K + 2 * half;
    const float* br0  = Bm + (size_t)(n_base + l16) * K + 2 * half;  // B stored [N,K]
    const float* br1  = br0 + (size_t)16 * K;
    const float* br2  = br0 + (size_t)32 * K;
    const float* br3  = br0 + (size_t)48 * K;

#define TITANS_FRAG(p, k) (*(const v2f*)((p) + (k)))   // 8B-aligned b64 load
#define TITANS_WMMA4()                                                               \
    do {                                                                             \
        acc[0] = __builtin_amdgcn_wmma_f32_16x16x4_f32(false, a, false, b0, (short)0, acc[0], false, false); \
        acc[1] = __builtin_amdgcn_wmma_f32_16x16x4_f32(false, a, false, b1, (short)0, acc[1], false, false); \
        acc[2] = __builtin_amdgcn_wmma_f32_16x16x4_f32(false, a, false, b2, (short)0, acc[2], false, false); \
        acc[3] = __builtin_amdgcn_wmma_f32_16x16x4_f32(false, a, false, b3, (short)0, acc[3], false, false); \
    } while (0)

    // software pipeline prologue: fragments for k=0
    v2f a  = TITANS_FRAG(arow, 0);
    v2f b0 = TITANS_FRAG(br0, 0);
    v2f b1 = TITANS_FRAG(br1, 0);
    v2f b2 = TITANS_FRAG(br2, 0);
    v2f b3 = TITANS_FRAG(br3, 0);

    // main loop: prefetch k0+4 (always in bounds), consume k0
#pragma unroll 4
    for (int k0 = 0; k0 < K - 4; k0 += 4) {
        const v2f an  = TITANS_FRAG(arow, k0 + 4);
        const v2f b0n = TITANS_FRAG(br0,  k0 + 4);
        const v2f b1n = TITANS_FRAG(br1,  k0 + 4);
        const v2f b2n = TITANS_FRAG(br2,  k0 + 4);
        const v2f b3n = TITANS_FRAG(br3,  k0 + 4);

        TITANS_WMMA4();

        a = an; b0 = b0n; b1 = b1n; b2 = b2n; b3 = b3n;
    }
    // peeled epilogue: consume the last fragments, no prefetch
    TITANS_WMMA4();

#undef TITANS_WMMA4
#undef TITANS_FRAG

    // C/D layout: VGPR r -> row = m_base + half*8 + r, col = tile_n_base + l16
#pragma unroll
    for (int nt = 0; nt < 4; ++nt) {
        const int col = n_base + nt * 16 + l16;
#pragma unroll
        for (int r = 0; r < 8; ++r) {
            const int row = m_base + half * 8 + r;
            float v = acc[nt][r];
            if (bias) v += bias[col];
            if (add)  v += add[(size_t)row * N + col];
            if (relu) v = fmaxf(v, 0.f);
            C[(size_t)row * N + col] = v;
        }
    }
}

// ---------------------------------------------------------------------------
// Row LayerNorm: Y[r,:] = (x - mu)/sqrt(var+eps)*g + b   (D = 256, block = 256)
// ---------------------------------------------------------------------------
__global__ __launch_bounds__(256)
void titans_ln_rows(const float* __restrict__ X, const float* __restrict__ g,
                    const float* __restrict__ b, float* __restrict__ Y)
{
    __shared__ float red[Dq];
    const int r = blockIdx.x, i = threadIdx.x;
    const float x = X[(size_t)r * Dq + i];

    red[i] = x; __syncthreads();
    for (int s = 128; s > 0; s >>= 1) { if (i < s) red[i] += red[i + s]; __syncthreads(); }
    const float mu = red[0] * (1.f / Dq);
    __syncthreads();

    const float d = x - mu;
    red[i] = d * d; __syncthreads();
    for (int s = 128; s > 0; s >>= 1) { if (i < s) red[i] += red[i + s]; __syncthreads(); }
    const float var = red[0] * (1.f / Dq);

    Y[(size_t)r * Dq + i] = d * rsqrtf(var + 1e-5f) * g[i] + b[i];
}

// ---------------------------------------------------------------------------
// In-place row L2 normalize: X[r,:] /= max(||X[r,:]||, 1e-12)
// ---------------------------------------------------------------------------
__global__ __launch_bounds__(256)
void titans_l2norm_rows(float* __restrict__ X)
{
    __shared__ float red[Dq];
    const int r = blockIdx.x, i = threadIdx.x;
    const float x = X[(size_t)r * Dq + i];
    red[i] = x * x; __syncthreads();
    for (int s = 128; s > 0; s >>= 1) { if (i < s) red[i] += red[i + s]; __syncthreads(); }
    const float inv = 1.f / fmaxf(sqrtf(red[0]), 1e-12f);
    X[(size_t)r * Dq + i] = x * inv;
}

// ---------------------------------------------------------------------------
// Gates: alpha[r] = sigmoid(X[r,:].Wa + ba); eta[r] = sigmoid(X[r,:].We + be)*D^-0.5
// ---------------------------------------------------------------------------
__global__ __launch_bounds__(256)
void titans_gates(const float* __restrict__ X,
                  const float* __restrict__ Wa, const float* __restrict__ ba,
                  const float* __restrict__ We, const float* __restrict__ be,
                  float* __restrict__ Al, float* __restrict__ Et)
{
    __shared__ float ra[Dq], re[Dq];
    const int r = blockIdx.x, i = threadIdx.x;
    const float x = X[(size_t)r * Dq + i];
    ra[i] = x * Wa[i];
    re[i] = x * We[i];
    __syncthreads();
    for (int s = 128; s > 0; s >>= 1) {
        if (i < s) { ra[i] += ra[i + s]; re[i] += re[i + s]; }
        __syncthreads();
    }
    if (i == 0) {
        Al[r] = 1.f / (1.f + expf(-(ra[0] + ba[0])));
        Et[r] = (1.f / (1.f + expf(-(re[0] + be[0])))) * 0.0625f;  // D^-0.5 = 1/16
    }
}

// ---------------------------------------------------------------------------
// Delta-rule scan. One WG per batch element; state [256x256] fp32 lives in LDS
// with row stride 257 (bank-conflict free on 64 banks). Thread i owns row i.
//   pred = S k ; err = v - pred ; S = (1-a) S + e * err k^T
// Then read = S q written to global.
// LDS: 256*257*4 + 256*4 = 264192 bytes (fits the 320 KB WGP LDS).
// state0 init uses gfx1250 async global->LDS DMA when available.
// ---------------------------------------------------------------------------
#define SCAN_STRIDE 257
#define SCAN_LDS (size_t)((Dq * SCAN_STRIDE + Dq) * sizeof(float))

__global__ __launch_bounds__(256)
void titans_scan(const float* __restrict__ state0,
                 const float* __restrict__ Km, const float* __restrict__ Vm,
                 const float* __restrict__ Al, const float* __restrict__ Et,
                 const float* __restrict__ Qm, float* __restrict__ Read)
{
    extern __shared__ float smem[];
    float* state = smem;                    // [256][257]
    float* kbuf  = smem + Dq * SCAN_STRIDE; // [256]

    const int b = blockIdx.x;
    const int i = threadIdx.x;              // row owned by this thread
    float* srow = state + i * SCAN_STRIDE;

    // load initial state row
    const float* s0 = state0 + ((size_t)b * Dq + i) * Dq;
#ifdef TITANS_ASYNC_LDS
    for (int j = 0; j < Dq; ++j)
        TITANS_ASYNC_CP_B32(s0 + j, srow + j);
    TITANS_WAIT_ASYNC();
#else
    for (int j = 0; j < Dq; ++j) srow[j] = s0[j];
#endif
    __syncthreads();

    for (int t = 0; t < Tq; ++t) {
        const size_t r = (size_t)b * Tq + t;
        kbuf[i]        = Km[r * Dq + i];
        const float v  = Vm[r * Dq + i];
        const float a  = Al[r];
        const float e  = Et[r];
        if (t + 1 < Tq) __builtin_prefetch(Km + (r + 1) * Dq + i, 0, 0);
        __syncthreads();

        float pred = 0.f;
        for (int j = 0; j < Dq; ++j) pred = fmaf(srow[j], kbuf[j], pred);

        const float oma = 1.f - a;
        const float ee  = e * (v - pred);
        for (int j = 0; j < Dq; ++j) srow[j] = oma * srow[j] + ee * kbuf[j];
        __syncthreads();  // protect kbuf reuse next step
    }

    // memory read with user query: read = S q
    kbuf[i] = Qm[(size_t)b * Dq + i];
    __syncthreads();
    float rd = 0.f;
    for (int j = 0; j < Dq; ++j) rd = fmaf(srow[j], kbuf[j], rd);
    Read[(size_t)b * Dq + i] = rd;
}

// ---------------------------------------------------------------------------
// Host launcher
// ---------------------------------------------------------------------------
extern "C" void kernel_launch(void* const* d_in, const int* in_sizes, int n_in,
                              void* d_out, int out_size, void* d_ws, size_t ws_size,
                              hipStream_t stream)
{
    const float* item_seq    = (const float*)d_in[0];   // [B,T,D]
    const float* user_static = (const float*)d_in[1];   // [B,D]
    const float* state0      = (const float*)d_in[2];   // [B,D,D]
    const float* Wq          = (const float*)d_in[3];
    const float* Wk          = (const float*)d_in[4];
    const float* Wv          = (const float*)d_in[5];
    const float* Wo          = (const float*)d_in[6];
    const float* bo          = (const float*)d_in[7];
    const float* Wa          = (const float*)d_in[8];
    const float* ba          = (const float*)d_in[9];
    const float* We          = (const float*)d_in[10];
    const float* be          = (const float*)d_in[11];
    const float* W1          = (const float*)d_in[12];  // [FF,D]
    const float* b1          = (const float*)d_in[13];
    const float* W2          = (const float*)d_in[14];  // [D,FF]
    const float* b2          = (const float*)d_in[15];
    const float* g1          = (const float*)d_in[16];
    const float* beta1       = (const float*)d_in[17];
    const float* g2          = (const float*)d_in[18];
    const float* beta2       = (const float*)d_in[19];
    float* out = (float*)d_out;

    // workspace layout (floats)
    float* ws = (float*)d_ws;
    const size_t BT = (size_t)Bq * Tq;             // 8192
    float* X   = ws;                               // [BT, D]
    float* Xu  = X  + BT * Dq;                     // [B, D]
    float* Km  = Xu + (size_t)Bq * Dq;             // [BT, D]
    float* Vm  = Km + BT * Dq;                     // [BT, D]
    float* Al  = Vm + BT * Dq;                     // [BT]
    float* Et  = Al + BT;                          // [BT]
    float* Qm  = Et + BT;                          // [B, D]
    float* Rd  = Qm + (size_t)Bq * Dq;             // [B, D]
    float* Xr  = Rd + (size_t)Bq * Dq;             // [B, D]  residual x
    float* X2  = Xr + (size_t)Bq * Dq;             // [B, D]  LN(x)
    float* Hm  = X2 + (size_t)Bq * Dq;             // [B, FF]
    (void)n_in; (void)in_sizes; (void)out_size; (void)ws_size;

    (void)hipFuncSetAttribute((const void*)titans_scan,
                              hipFuncAttributeMaxDynamicSharedMemorySize,
                              (int)SCAN_LDS);

    const dim3 blk(256);

    // Phase 1: state-independent projections (all WMMA fp32)
    titans_ln_rows<<<(int)BT, 256, 0, stream>>>(item_seq, g1, beta1, X);
    titans_ln_rows<<<Bq, 256, 0, stream>>>(user_static, g1, beta1, Xu);

    titans_gemm_nt_f32_wmma<<<dim3(Dq/64, BT/128), blk, 0, stream>>>(
        X, Wk, Km, nullptr, nullptr, (int)BT, Dq, Dq, 0);
    titans_gemm_nt_f32_wmma<<<dim3(Dq/64, BT/128), blk, 0, stream>>>(
        X, Wv, Vm, nullptr, nullptr, (int)BT, Dq, Dq, 0);
    titans_gemm_nt_f32_wmma<<<dim3(Dq/64, Bq/128), blk, 0, stream>>>(
        Xu, Wq, Qm, nullptr, nullptr, Bq, Dq, Dq, 0);

    titans_gates<<<(int)BT, 256, 0, stream>>>(X, Wa, ba, We, be, Al, Et);
    titans_l2norm_rows<<<(int)BT, 256, 0, stream>>>(Km);
    titans_l2norm_rows<<<Bq, 256, 0, stream>>>(Qm);

    // Phase 2: serial delta-rule scan, LDS-resident state, fused read
    titans_scan<<<Bq, 256, SCAN_LDS, stream>>>(state0, Km, Vm, Al, Et, Qm, Rd);

    // Phase 3: output projection + FFN (WMMA, fused bias/residual/ReLU)
    titans_gemm_nt_f32_wmma<<<dim3(Dq/64, Bq/128), blk, 0, stream>>>(
        Rd, Wo, Xr, bo, user_static, Bq, Dq, Dq, 0);         // x = us + read@Wo^T + bo
    titans_ln_rows<<<Bq, 256, 0, stream>>>(Xr, g2, beta2, X2);
    titans_gemm_nt_f32_wmma<<<dim3(FFq/64, Bq/128), blk, 0, stream>>>(
        X2, W1, Hm, b1, nullptr, Bq, FFq, Dq, 1);            // H = relu(x2@W1^T + b1)
    titans_gemm_nt_f32_wmma<<<dim3(Dq/64, Bq/128), blk, 0, stream>>>(
        Hm, W2, out, b2, Xr, Bq, Dq, FFq, 0);                // out = x + H@W2^T + b2
}